// TreeCRFLayer_35115652612645
// MI455X (gfx1250) — compile-verified
//
#include <hip/hip_runtime.h>
#include <stdint.h>

// Tree-CRF belief propagation, full binary tree L=1023, C=2, B=4096.
// One block = 4 batch rows; whole tree state lives in LDS.
// X tile staged via CDNA5 async global->LDS (ASYNCcnt) path.

#define NB       4          // batch rows per block
#define LNODES   1023       // full binary tree, depths 0..9
#define ROWF     (2 * LNODES)   // floats per batch row (2046)
#define THREADS  256        // 8 waves of 32

__device__ __forceinline__ float lse2(float a, float b) {
    float m = fmaxf(a, b);
    float n = fminf(a, b);
    return m + log1pf(expf(n - m));
}

__global__ __launch_bounds__(THREADS)
void treecrf_kernel(const float* __restrict__ X,
                    const float* __restrict__ pairs,
                    float* __restrict__ out,
                    int Btot) {
    // Xs: staged X for NB rows, layout [b][node*2+c]; overwritten in the
    //     down pass with S = X + beta (in place, level by level).
    // A : alpha, layout [node][b] as float2 (classes packed).
    __shared__ __align__(16) float  Xs[NB * ROWF];   // 32736 B
    __shared__ __align__(8)  float2 A[LNODES * NB];  // 32736 B

    const int tid = threadIdx.x;
    const int b0  = blockIdx.x * NB;
    const int vb  = min(NB, Btot - b0);              // valid rows in this tile
    const float* src = X + (size_t)b0 * ROWF;
    float*       dst = out + (size_t)b0 * ROWF;

    // ---- Stage X tile into LDS ----
    if (vb == NB) {
        // 4 adjacent rows form one contiguous 32736-B range: async b128 copy.
        uint32_t lds_base = (uint32_t)(uintptr_t)(&Xs[0]);
        const int n128 = (NB * ROWF * 4) / 16;       // 2046 16-byte chunks
        for (int i = tid; i < n128; i += THREADS) {
            uint32_t laddr = lds_base + (uint32_t)(i * 16);
            uint64_t gaddr = (uint64_t)(uintptr_t)src + (uint64_t)(i * 16);
            asm volatile("global_load_async_to_lds_b128 %0, %1, off"
                         :: "v"(laddr), "v"(gaddr)
                         : "memory");
        }
    } else {
        // Tail tile (not hit for B=4096): plain loads + zero-fill invalid rows.
        for (int i = tid; i < NB * ROWF; i += THREADS) {
            int b = i / ROWF;
            Xs[i] = (b < vb) ? src[i] : 0.0f;
        }
    }

    // Zero leaf alphas (nodes 511..1022) while the async copy is in flight.
    for (int w = tid; w < 512 * NB; w += THREADS) {
        int node = 511 + (w >> 2);
        int b    = w & (NB - 1);
        A[node * NB + b] = make_float2(0.0f, 0.0f);
    }

#if __has_builtin(__builtin_amdgcn_s_wait_asynccnt)
    __builtin_amdgcn_s_wait_asynccnt(0);
#else
    asm volatile("s_wait_asynccnt 0" ::: "memory");
#endif
    __syncthreads();

    // ---- Upward pass (leaves -> root), parallel over parents ----
    for (int d = 9; d >= 1; --d) {
        const int np = 1 << (d - 1);     // parents at depth d-1
        const int fp = np - 1;           // first parent index
        for (int w = tid; w < np * NB; w += THREADS) {
            int b  = w & (NB - 1);
            int p  = fp + (w >> 2);
            int cl = 2 * p + 1;
            int cr = cl + 1;
            float2 al = A[cl * NB + b];
            float2 ar = A[cr * NB + b];
            const float* xb = &Xs[b * ROWF];
            float l0 = xb[2 * cl + 0] + al.x;
            float l1 = xb[2 * cl + 1] + al.y;
            float r0 = xb[2 * cr + 0] + ar.x;
            float r1 = xb[2 * cr + 1] + ar.y;
            const float* PL = pairs + ((size_t)p * LNODES + cl) * 4; // [yi][yj]
            const float* PR = pairs + ((size_t)p * LNODES + cr) * 4;
            float m0 = lse2(l0 + PL[0], l1 + PL[1]) + lse2(r0 + PR[0], r1 + PR[1]);
            float m1 = lse2(l0 + PL[2], l1 + PL[3]) + lse2(r0 + PR[2], r1 + PR[3]);
            A[p * NB + b] = make_float2(m0, m1);
        }
        __syncthreads();
    }

    // ---- Root output (beta = 0; Xs[root] already equals S[root]) ----
    if (tid < NB) {
        int b = tid;
        float2 a = A[0 + b];
        float t0 = Xs[b * ROWF + 0] + a.x;
        float t1 = Xs[b * ROWF + 1] + a.y;
        float Z  = lse2(t0, t1);
        if (b < vb) {
            dst[b * ROWF + 0] = t0 - Z;
            dst[b * ROWF + 1] = t1 - Z;
        }
    }

    // ---- Downward pass (root -> leaves), parallel over nodes ----
    // Invariant entering level d: Xs[j] holds S[j] = X[j] + beta[j] for all
    // nodes j at depth < d. Each node reads its parent's S, emits its
    // normalized log-marginal, then converts its own Xs slot to S.
    for (int d = 1; d <= 9; ++d) {
        const int nn = 1 << d;
        const int fn = nn - 1;
        for (int w = tid; w < nn * NB; w += THREADS) {
            int b = w & (NB - 1);
            int i = fn + (w >> 2);
            int j = (i - 1) >> 1;
            float* xb = &Xs[b * ROWF];
            float Sj0 = xb[2 * j + 0];
            float Sj1 = xb[2 * j + 1];
            const float* P = pairs + ((size_t)i * LNODES + j) * 4; // pairs[child, parent]
            float m0 = lse2(Sj0 + P[0], Sj1 + P[1]);
            float m1 = lse2(Sj0 + P[2], Sj1 + P[3]);
            float x0 = xb[2 * i + 0];
            float x1 = xb[2 * i + 1];
            float2 a = A[i * NB + b];
            float t0 = x0 + a.x + m0;
            float t1 = x1 + a.y + m1;
            float Z  = lse2(t0, t1);
            if (b < vb) {
                dst[b * ROWF + 2 * i + 0] = t0 - Z;
                dst[b * ROWF + 2 * i + 1] = t1 - Z;
            }
            xb[2 * i + 0] = x0 + m0;   // S[i] for next level's children
            xb[2 * i + 1] = x1 + m1;
        }
        __syncthreads();
    }
}

extern "C" void kernel_launch(void* const* d_in, const int* in_sizes, int n_in,
                              void* d_out, int out_size, void* d_ws, size_t ws_size,
                              hipStream_t stream) {
    const float* X     = (const float*)d_in[0];   // (B, 1023, 2) fp32
    const float* pairs = (const float*)d_in[1];   // (1023, 1023, 2, 2) fp32
    float*       out   = (float*)d_out;           // (B, 1023, 2) fp32
    (void)d_ws; (void)ws_size; (void)n_in; (void)out_size;

    int Btot = in_sizes[0] / ROWF;                // 4096
    int grid = (Btot + NB - 1) / NB;              // 1024 blocks
    treecrf_kernel<<<grid, THREADS, 0, stream>>>(X, pairs, out, Btot);
}